// DSNN_47768626266261
// MI455X (gfx1250) — compile-verified
//
#include <hip/hip_runtime.h>

#define ALPHA 0.9f
#define BETA  0.85f
#define SIM_TIME 50

typedef float v2f __attribute__((ext_vector_type(2)));
typedef float v8f __attribute__((ext_vector_type(8)));
typedef int   v4i __attribute__((vector_size(4 * sizeof(int))));  // 128-bit payload type

#define BM 32
#define BN 64
#define BK 32
#define APAD 36   // dword pitch: rows 16B-aligned, conflict-free frag gathers
#define BPAD 72

#if __has_builtin(__builtin_amdgcn_global_load_async_to_lds_b128) && \
    __has_builtin(__builtin_amdgcn_s_wait_asynccnt)
#define USE_ASYNC_LDS 1
#else
#define USE_ASYNC_LDS 0
#warning "gfx1250 async-to-LDS builtins unavailable on this toolchain; using sync LDS staging"
#endif

// Builtin signature (from clang diagnostic): (v4i AS1* src, v4i AS3* dst, imm off, imm cpol)
#define AS_GLOBAL_V4I(p) ((__attribute__((address_space(1))) v4i*)(p))
#define AS_LDS_V4I(p)    ((__attribute__((address_space(3))) v4i*)(p))

// Fused GEMM + SNN state-update epilogue.
// MODE 0: Cout = A@B                                        (H0 precompute)
// MODE 1: h=A@B; s=ALPHA*syn+h; syn=s; m=BETA*mem+s;
//         spk=(m-1>0); Cout=spk; mem=spk?0:m                (hidden layer)
// MODE 2: h=A@B; s=ALPHA*syn+h; syn=s; mem=BETA*mem+s       (output layer, no reset)
template <int MODE>
__global__ __launch_bounds__(256) void snn_gemm(
    const float* __restrict__ A,   // M x K row-major (spikes / inputs)
    const float* __restrict__ B,   // K x N row-major (weights)
    float* __restrict__ Cout,      // M x N (H0 or spike out; unused MODE 2)
    float* __restrict__ syn,       // M x N state (unused MODE 0)
    float* __restrict__ mem,       // M x N state (unused MODE 0)
    int M, int N, int K)
{
    __shared__ float Alds[2][BM][APAD];   // double-buffered K-blocks
    __shared__ float Blds[2][BK][BPAD];

    const int tid  = threadIdx.x;
    const int lane = tid & 31;
    const int wid  = tid >> 5;
    const int half = lane >> 4;
    const int l15  = lane & 15;

    const int m0 = blockIdx.y * BM;
    const int n0 = blockIdx.x * BN;
    const int m_w = (wid >> 2) * 16;
    const int n_w = (wid & 3) * 16;

    // Cooperative staging map: 1 x b128 of A + 2 x b128 of B per thread.
    const int a_row = tid >> 3;        // 0..31
    const int a_col = (tid & 7) * 4;   // 0..28
    const int b_row = tid >> 4;        // 0..15
    const int b_col = (tid & 15) * 4;  // 0..60

    // Pointer-increment addressing (no in-loop 64-bit index rebuilds).
    const float* Ag = A + (size_t)(m0 + a_row) * K + a_col;
    const float* Bg = B + (size_t)b_row * N + n0 + b_col;
    const size_t bStep = (size_t)BK * N;
    const size_t bHalf = (size_t)16 * N;

    v8f acc = {};

    // 8 x V_WMMA_F32_16X16X4_F32 per K-block.
    // A frag: lane -> A[m_w+l15][kk+2*half +{0,1}] (contiguous, ds_load_b64)
    // B frag: lane -> B[kk+2*half +{0,1}][n_w+l15]
    auto compute = [&](int buf) {
#pragma unroll
        for (int kk = 0; kk < BK; kk += 4) {
            v2f a = *(const v2f*)&Alds[buf][m_w + l15][kk + 2 * half];
            v2f b;
            b.x = Blds[buf][kk + 2 * half][n_w + l15];
            b.y = Blds[buf][kk + 2 * half + 1][n_w + l15];
            acc = __builtin_amdgcn_wmma_f32_16x16x4_f32(
                false, a, false, b, (short)0, acc, false, false);
        }
    };

    const int nkb = K / BK;

#if USE_ASYNC_LDS
    // Async DMA staging: global -> LDS with no VGPR round-trip (ASYNCcnt-tracked).
    auto stage = [&](int buf) {
        __builtin_amdgcn_global_load_async_to_lds_b128(
            AS_GLOBAL_V4I(Ag), AS_LDS_V4I(&Alds[buf][a_row][a_col]), 0, 0);
        __builtin_amdgcn_global_load_async_to_lds_b128(
            AS_GLOBAL_V4I(Bg), AS_LDS_V4I(&Blds[buf][b_row][b_col]), 0, 0);
        __builtin_amdgcn_global_load_async_to_lds_b128(
            AS_GLOBAL_V4I(Bg + bHalf), AS_LDS_V4I(&Blds[buf][b_row + 16][b_col]), 0, 0);
        Ag += BK;
        Bg += bStep;
    };

    stage(0);
    for (int kb = 0; kb < nkb - 1; ++kb) {
        const int cur = kb & 1;
        __builtin_amdgcn_s_wait_asynccnt(0);  // our block-kb pieces landed in LDS
        __syncthreads();                      // everyone's pieces landed; WAR fence
        stage(cur ^ 1);                       // next block DMAs overlap the WMMAs
        compute(cur);
    }
    __builtin_amdgcn_s_wait_asynccnt(0);
    __syncthreads();
    compute((nkb - 1) & 1);
#else
    // Sync fallback: global->VGPR->LDS, double-buffered, one barrier/iteration.
    float4 aR  = *(const float4*)Ag;
    float4 bR0 = *(const float4*)Bg;
    float4 bR1 = *(const float4*)(Bg + bHalf);
    for (int kb = 0; kb < nkb - 1; ++kb) {
        const int cur = kb & 1;
        *(float4*)&Alds[cur][a_row][a_col]      = aR;
        *(float4*)&Blds[cur][b_row][b_col]      = bR0;
        *(float4*)&Blds[cur][b_row + 16][b_col] = bR1;
        __syncthreads();
        Ag += BK;
        Bg += bStep;
        aR  = *(const float4*)Ag;
        bR0 = *(const float4*)Bg;
        bR1 = *(const float4*)(Bg + bHalf);
        __builtin_prefetch(Ag + BK, 0, 3);            // speculative; OOB drops
        __builtin_prefetch(Bg + bStep, 0, 3);
        __builtin_prefetch(Bg + bStep + bHalf, 0, 3);
        compute(cur);
    }
    {
        const int cur = (nkb - 1) & 1;
        *(float4*)&Alds[cur][a_row][a_col]      = aR;
        *(float4*)&Blds[cur][b_row][b_col]      = bR0;
        *(float4*)&Blds[cur][b_row + 16][b_col] = bR1;
        __syncthreads();
        compute(cur);
    }
#endif

    // Epilogue: C layout -> VGPR r, lane l is row r + 8*(l>=16), col l&15.
    const int col = n0 + n_w + l15;
#pragma unroll
    for (int r = 0; r < 8; ++r) {
        const int row    = m0 + m_w + 8 * half + r;
        const size_t idx = (size_t)row * N + col;
        const float h    = acc[r];
        if (MODE == 0) {
            Cout[idx] = h;
        } else {
            const float s = ALPHA * syn[idx] + h;
            syn[idx] = s;
            const float m = BETA * mem[idx] + s;
            if (MODE == 1) {
                const bool sp = (m - 1.0f) > 0.0f;
                Cout[idx] = sp ? 1.0f : 0.0f;
                mem[idx]  = sp ? 0.0f : m;
            } else {
                mem[idx] = m;  // output layer never resets
            }
        }
    }
}

// Layer 0 per-timestep update: drive H0 is time-invariant; layer 0 has no syn decay.
__global__ __launch_bounds__(256) void l0_update(
    const float* __restrict__ H0, float* __restrict__ mem0,
    float* __restrict__ spk0, int n)
{
    const int i = blockIdx.x * blockDim.x + threadIdx.x;
    if (i < n) {
        const float m = BETA * mem0[i] + H0[i];
        const bool sp = (m - 1.0f) > 0.0f;
        spk0[i] = sp ? 1.0f : 0.0f;
        mem0[i] = sp ? 0.0f : m;
    }
}

__global__ __launch_bounds__(256) void zero_f32(float* __restrict__ p, size_t n)
{
    const size_t i = (size_t)blockIdx.x * blockDim.x + threadIdx.x;
    if (i < n) p[i] = 0.0f;
}

extern "C" void kernel_launch(void* const* d_in, const int* in_sizes, int n_in,
                              void* d_out, int out_size, void* d_ws, size_t ws_size,
                              hipStream_t stream)
{
    (void)in_sizes; (void)n_in; (void)out_size; (void)ws_size;

    const float* inputs = (const float*)d_in[0];  // 128 x 1024
    const float* W0     = (const float*)d_in[1];  // 1024 x 2048
    const float* W1     = (const float*)d_in[2];  // 2048 x 2048
    const float* W2     = (const float*)d_in[3];  // 2048 x 512
    float* out          = (float*)d_out;          // 128 x 512 == mem2

    const int Bt = 128, N0 = 2048, N1 = 2048, N2 = 512, K0 = 1024, K12 = 2048;
    const size_t big   = (size_t)Bt * 2048;  // 262144
    const size_t small = (size_t)Bt * 512;   // 65536

    // Workspace (fp32): [H0 | spk0 | spk1 | mem0 syn1 mem1 syn2]
    float* ws   = (float*)d_ws;
    float* H0   = ws;
    float* spk0 = H0 + big;
    float* spk1 = spk0 + big;
    float* mem0 = spk1 + big;   // start of zero-init state region
    float* syn1 = mem0 + big;
    float* mem1 = syn1 + big;
    float* syn2 = mem1 + big;

    // Re-zero persistent state every call (ws/d_out arrive poisoned).
    const size_t zn = 3 * big + small;  // mem0, syn1, mem1, syn2 (contiguous)
    zero_f32<<<(unsigned)((zn + 255) / 256), 256, 0, stream>>>(mem0, zn);
    zero_f32<<<(unsigned)((small + 255) / 256), 256, 0, stream>>>(out, small);

    // Hoisted time-invariant drive: H0 = inputs @ W0 (done once, not 50x).
    snn_gemm<0><<<dim3(N0 / BN, Bt / BM), 256, 0, stream>>>(
        inputs, W0, H0, nullptr, nullptr, Bt, N0, K0);

    for (int t = 0; t < SIM_TIME; ++t) {
        l0_update<<<(unsigned)(big / 256), 256, 0, stream>>>(H0, mem0, spk0, (int)big);
        snn_gemm<1><<<dim3(N1 / BN, Bt / BM), 256, 0, stream>>>(
            spk0, W1, spk1, syn1, mem1, Bt, N1, K12);
        snn_gemm<2><<<dim3(N2 / BN, Bt / BM), 256, 0, stream>>>(
            spk1, W2, nullptr, syn2, out, Bt, N2, K12);
    }
}